// augGNN_27994596835783
// MI455X (gfx1250) — compile-verified
//
#include <hip/hip_runtime.h>
#include <math.h>

typedef __attribute__((ext_vector_type(2))) float f32x2;
typedef __attribute__((ext_vector_type(8))) float f32x8;

#define BN_EPS 1e-5f

__device__ __forceinline__ f32x8 wmma4(f32x2 a, f32x2 b, f32x8 c) {
  // V_WMMA_F32_16X16X4_F32 : D = A(16x4) x B(4x16) + C(16x16)
  return __builtin_amdgcn_wmma_f32_16x16x4_f32(false, a, false, b, (short)0, c,
                                               false, false);
}

// gfx1250 async global->LDS copy (ASYNCcnt-tracked). Low 32 bits of a flat
// shared pointer are the wave-relative LDS byte address (aperture mapping
// discards the upper bits), which is what VDST expects.
__device__ __forceinline__ void async_copy_b128(void* lds_ptr, const void* gptr) {
  uint32_t l = (uint32_t)(uintptr_t)lds_ptr;
  uint64_t g = (uint64_t)(uintptr_t)gptr;
  asm volatile("global_load_async_to_lds_b128 %0, %1, off" ::"v"(l), "v"(g)
               : "memory");
}
__device__ __forceinline__ void wait_async0() {
  asm volatile("s_wait_asynccnt 0x0" ::: "memory");
}

// ---------------------------------------------------------------------------
// Kernel 0: fold BN params into affine / scaled weights
// ---------------------------------------------------------------------------
__global__ void prep_kernel(const float* __restrict__ w1, const float* __restrict__ b1,
                            const float* __restrict__ g1, const float* __restrict__ be1,
                            const float* __restrict__ m1, const float* __restrict__ v1,
                            const float* __restrict__ w3, const float* __restrict__ b3,
                            const float* __restrict__ g2, const float* __restrict__ be2,
                            const float* __restrict__ m2, const float* __restrict__ v2,
                            float* __restrict__ alpha, float* __restrict__ beta,
                            float* __restrict__ w3s, float* __restrict__ c3) {
  int t = threadIdx.x;
  if (t < 256) {
    float rs = rsqrtf(v1[t] + BN_EPS);
    float sc = g1[t] * rs;
    alpha[t] = w1[t] * sc;
    beta[t] = (b1[t] - m1[t]) * sc + be1[t];
  }
  if (t < 64) {
    float rs = rsqrtf(v2[t] + BN_EPS);
    c3[t] = (b3[t] - m2[t]) * g2[t] * rs + be2[t];
  }
  for (int idx = t; idx < 128 * 64; idx += blockDim.x) {
    int j = idx & 63;
    w3s[idx] = w3[idx] * g2[j] * rsqrtf(v2[j] + BN_EPS);
  }
}

// a0[i] = x[i,0]; a1[i] = x[i,1]
__global__ void init_a_kernel(const float* __restrict__ x, float* __restrict__ a0,
                              float* __restrict__ a1, int n) {
  int i = blockIdx.x * blockDim.x + threadIdx.x;
  if (i < n) {
    a0[i] = x[2 * i];
    a1[i] = x[2 * i + 1];
  }
}

// scalar-column scatter-sum over edges for both columns at once
__global__ void agg_scalar_kernel(const long long* __restrict__ ei,
                                  const float* __restrict__ x, float* __restrict__ a0,
                                  float* __restrict__ a1, int E) {
  int e = blockIdx.x * blockDim.x + threadIdx.x;
  if (e >= E) return;
  long long s = ei[e];
  long long d = ei[(long long)E + e];
  atomicAdd(&a0[d], x[2 * s]);
  atomicAdd(&a1[d], x[2 * s + 1]);
}

// ---------------------------------------------------------------------------
// MLP1: p[n,128] = relu(a*alpha + beta)[n,256] @ w2[256,128] + b2
// block = 256 threads (8 waves); each wave owns one 16-wide output col tile;
// block covers one 16-node tile; K=256 consumed 4 at a time via f32 WMMA.
// ---------------------------------------------------------------------------
__global__ void mlp1_kernel(const float* __restrict__ a, const float* __restrict__ alpha,
                            const float* __restrict__ beta, const float* __restrict__ w2,
                            const float* __restrict__ b2, float* __restrict__ p, int n) {
  __shared__ float als[256];
  __shared__ float bes[256];
  if (threadIdx.x < 256) {
    als[threadIdx.x] = alpha[threadIdx.x];
    bes[threadIdx.x] = beta[threadIdx.x];
  }
  __syncthreads();

  int lane = threadIdx.x & 31;
  int wave = threadIdx.x >> 5;  // 0..7
  int m = lane & 15;            // A row / B col within tile
  int g = lane >> 4;            // K-pair group
  int colbase = wave * 16;
  int tile = blockIdx.x;

  int row_m = tile * 16 + m;
  float aval = (row_m < n) ? a[row_m] : 0.0f;

  f32x8 acc = {};
  for (int t = 0; t < 64; ++t) {
    int k0 = 4 * t + 2 * g;
    f32x2 av, bv;
    av.x = fmaxf(fmaf(aval, als[k0], bes[k0]), 0.0f);
    av.y = fmaxf(fmaf(aval, als[k0 + 1], bes[k0 + 1]), 0.0f);
    bv.x = w2[k0 * 128 + colbase + m];
    bv.y = w2[(k0 + 1) * 128 + colbase + m];
    acc = wmma4(av, bv, acc);
  }

  int col = colbase + m;
  float bias = b2[col];
  if (tile * 16 + 16 <= n) {  // full tile: uniform fast path, no exec churn
    float* dst = p + (size_t)(tile * 16 + 8 * g) * 128 + col;
#pragma unroll
    for (int r = 0; r < 8; ++r) dst[(size_t)r * 128] = acc[r] + bias;
  } else {
#pragma unroll
    for (int r = 0; r < 8; ++r) {
      int row = tile * 16 + r + 8 * g;
      if (row < n) p[(long long)row * 128 + col] = acc[r] + bias;
    }
  }
}

// 128-wide feature scatter-add: q[dst] += p[src] (q pre-initialized to p)
__global__ void agg128_kernel(const long long* __restrict__ ei,
                              const float* __restrict__ p, float* __restrict__ q, int E) {
  int gid = blockIdx.x * blockDim.x + threadIdx.x;
  int e = gid >> 5;
  if (e >= E) return;
  int fg = (gid & 31) * 4;
  long long s = ei[e];
  long long d = ei[(long long)E + e];
  const float4 v = *(const float4*)(p + (long long)s * 128 + fg);
  float* dq = q + (long long)d * 128 + fg;
  atomicAdd(dq + 0, v.x);
  atomicAdd(dq + 1, v.y);
  atomicAdd(dq + 2, v.z);
  atomicAdd(dq + 3, v.w);
}

// ---------------------------------------------------------------------------
// MLP2: h[n,64] = relu(q[n,128] @ w3s[128,64] + c3)
// block = 128 threads (4 waves = 4 col tiles), one 16-node tile per block.
// ---------------------------------------------------------------------------
__global__ void mlp2_kernel(const float* __restrict__ q, const float* __restrict__ w3s,
                            const float* __restrict__ c3, float* __restrict__ h, int n) {
  __shared__ float qs[16 * 128];
  int tile = blockIdx.x;
  for (int idx = threadIdx.x; idx < 16 * 128; idx += blockDim.x) {
    int r = idx >> 7, c = idx & 127;
    int row = tile * 16 + r;
    qs[idx] = (row < n) ? q[(long long)row * 128 + c] : 0.0f;
  }
  __syncthreads();

  int lane = threadIdx.x & 31;
  int wave = threadIdx.x >> 5;  // 0..3
  int m = lane & 15;
  int g = lane >> 4;
  int colbase = wave * 16;

  f32x8 acc = {};
  for (int t = 0; t < 32; ++t) {
    int k0 = 4 * t + 2 * g;
    f32x2 av, bv;
    av.x = qs[m * 128 + k0];
    av.y = qs[m * 128 + k0 + 1];
    bv.x = w3s[k0 * 64 + colbase + m];
    bv.y = w3s[(k0 + 1) * 64 + colbase + m];
    acc = wmma4(av, bv, acc);
  }

  int col = colbase + m;
  float cc = c3[col];
  if (tile * 16 + 16 <= n) {
    float* dst = h + (size_t)(tile * 16 + 8 * g) * 64 + col;
#pragma unroll
    for (int r = 0; r < 8; ++r) dst[(size_t)r * 64] = fmaxf(acc[r] + cc, 0.0f);
  } else {
#pragma unroll
    for (int r = 0; r < 8; ++r) {
      int row = tile * 16 + r + 8 * g;
      if (row < n) h[(long long)row * 64 + col] = fmaxf(acc[r] + cc, 0.0f);
    }
  }
}

// ---------------------------------------------------------------------------
// Bilinear + head.
// hb[n,k] = sum_{i,j} h1[n,i]*wb[k,i,j]*h2[n,j] + bb[k]
//         = G[n,:] @ WBt[:,k],  G[n,i*64+j]=h1[n,i]*h2[n,j],
//           WBt column k = wb[k] flattened (contiguous).
// block = 192 threads (6 waves = 6 k-col tiles of 16 -> 96), 64 nodes/block
// (4 node tiles per wave). wb chunks are staged with ASYNC global->LDS b128
// copies. Then per-node head: relu(hb@w5+b5)@w6+b6 -> log_softmax.
// ---------------------------------------------------------------------------
#define KC 32
__global__ void bilinear_head_kernel(const float* __restrict__ h1,
                                     const float* __restrict__ h2,
                                     const float* __restrict__ wb,
                                     const float* __restrict__ bb,
                                     const float* __restrict__ w5,
                                     const float* __restrict__ b5,
                                     const float* __restrict__ w6,
                                     const float* __restrict__ b6,
                                     float* __restrict__ out, int n) {
  // 8192 floats for h1/h2 tiles + 6144-float union region (wbs chunk / hbs)
  __shared__ float smem[8192 + 6144];
  float* h1s = smem;         // [64][64]
  float* h2s = smem + 4096;  // [64][64]
  float* wbs = smem + 8192;  // [96][KC]  (3072 floats, lives inside union)
  float* hbs = smem + 8192;  // [64][96]  (6144 floats, reused after K loop)

  int nodebase = blockIdx.x * 64;
  if (nodebase + 64 <= n) {  // full 64-node block: vectorized stage
    const float4* s1 = (const float4*)(h1 + (size_t)nodebase * 64);
    const float4* s2 = (const float4*)(h2 + (size_t)nodebase * 64);
    float4* d1 = (float4*)h1s;
    float4* d2 = (float4*)h2s;
    for (int i = threadIdx.x; i < 1024; i += blockDim.x) {
      d1[i] = s1[i];
      d2[i] = s2[i];
    }
  } else {
    for (int idx = threadIdx.x; idx < 4096; idx += blockDim.x) {
      int r = idx >> 6, c = idx & 63;
      int row = nodebase + r;
      bool ok = row < n;
      h1s[idx] = ok ? h1[(long long)row * 64 + c] : 0.0f;
      h2s[idx] = ok ? h2[(long long)row * 64 + c] : 0.0f;
    }
  }
  __syncthreads();

  int lane = threadIdx.x & 31;
  int wave = threadIdx.x >> 5;  // 0..5
  int m = lane & 15;
  int g = lane >> 4;
  int colbase = wave * 16;

  f32x8 acc[4] = {f32x8{}, f32x8{}, f32x8{}, f32x8{}};

  for (int kc = 0; kc < 4096; kc += KC) {
    // stage wb chunk via async global->LDS b128 copies (always in-bounds,
    // 16B aligned: kc%32==0, kk%4==0). 768 float4's / 192 threads = 4 each.
    for (int f4 = threadIdx.x; f4 < (96 * KC) / 4; f4 += blockDim.x) {
      int flat = f4 * 4;
      int c = flat / KC, kk = flat - c * KC;
      async_copy_b128(&wbs[flat], wb + (size_t)c * 4096 + kc + kk);
    }
    wait_async0();
    __syncthreads();
#pragma unroll
    for (int t = 0; t < KC / 4; ++t) {
      int kl = 4 * t + 2 * g;
      int kglob = kc + kl;
      int i = kglob >> 6;
      int j0 = kglob & 63;  // even, j0+1 stays in same i row
      f32x2 bv;
      bv.x = wbs[(colbase + m) * KC + kl];
      bv.y = wbs[(colbase + m) * KC + kl + 1];
#pragma unroll
      for (int nt = 0; nt < 4; ++nt) {
        int rl = (nt * 16 + m) * 64;
        float h1v = h1s[rl + i];
        f32x2 av;
        av.x = h1v * h2s[rl + j0];
        av.y = h1v * h2s[rl + j0 + 1];
        acc[nt] = wmma4(av, bv, acc[nt]);
      }
    }
    __syncthreads();  // protect wbs region before next stage / hbs reuse
  }

  int col = colbase + m;
  float bias = bb[col];
#pragma unroll
  for (int nt = 0; nt < 4; ++nt)
#pragma unroll
    for (int r = 0; r < 8; ++r)
      hbs[(nt * 16 + r + 8 * g) * 96 + col] = acc[nt][r] + bias;
  __syncthreads();

  // head: one thread per node (first 64 threads)
  if (threadIdx.x < 64) {
    int node = nodebase + threadIdx.x;
    if (node < n) {
      const float* hv = &hbs[threadIdx.x * 96];
      float y[6];
#pragma unroll
      for (int o = 0; o < 6; ++o) {
        float av = b5[o];
        for (int k = 0; k < 96; ++k) av = fmaf(hv[k], w5[k * 6 + o], av);
        y[o] = fmaxf(av, 0.0f);
      }
      float z[6];
      float mx = -3.4e38f;
#pragma unroll
      for (int o = 0; o < 6; ++o) {
        float av = b6[o];
#pragma unroll
        for (int k = 0; k < 6; ++k) av = fmaf(y[k], w6[k * 6 + o], av);
        z[o] = av;
        mx = fmaxf(mx, av);
      }
      float s = 0.0f;
#pragma unroll
      for (int o = 0; o < 6; ++o) s += expf(z[o] - mx);
      float ls = mx + logf(s);
#pragma unroll
      for (int o = 0; o < 6; ++o) out[(long long)node * 6 + o] = z[o] - ls;
    }
  }
}

// ---------------------------------------------------------------------------
extern "C" void kernel_launch(void* const* d_in, const int* in_sizes, int n_in,
                              void* d_out, int out_size, void* d_ws, size_t ws_size,
                              hipStream_t stream) {
  const float* x = (const float*)d_in[0];
  const long long* ei = (const long long*)d_in[1];
  const float* w1 = (const float*)d_in[2];
  const float* b1 = (const float*)d_in[3];
  const float* g1 = (const float*)d_in[4];
  const float* be1 = (const float*)d_in[5];
  const float* m1 = (const float*)d_in[6];
  const float* v1 = (const float*)d_in[7];
  const float* w2 = (const float*)d_in[8];
  const float* b2 = (const float*)d_in[9];
  const float* w3 = (const float*)d_in[10];
  const float* b3 = (const float*)d_in[11];
  const float* g2 = (const float*)d_in[12];
  const float* be2 = (const float*)d_in[13];
  const float* m2 = (const float*)d_in[14];
  const float* v2 = (const float*)d_in[15];
  const float* wb = (const float*)d_in[16];
  const float* bb = (const float*)d_in[17];
  const float* w5 = (const float*)d_in[18];
  const float* b5 = (const float*)d_in[19];
  const float* w6 = (const float*)d_in[20];
  const float* b6 = (const float*)d_in[21];

  int N = in_sizes[0] / 2;
  int E = in_sizes[1] / 2;

  float* ws = (float*)d_ws;
  size_t off = 0;
  float* A0 = ws + off;    off += (size_t)N;
  float* A1 = ws + off;    off += (size_t)N;
  float* ALPHA = ws + off; off += 256;
  float* BETA = ws + off;  off += 256;
  float* W3S = ws + off;   off += 128 * 64;
  float* C3 = ws + off;    off += 64;
  off = (off + 15) & ~(size_t)15;  // 16-float align for float4 loads
  float* P = ws + off;     off += (size_t)N * 128;
  float* Q = ws + off;     off += (size_t)N * 128;
  float* H1 = ws + off;    off += (size_t)N * 64;
  float* H2 = ws + off;    off += (size_t)N * 64;

  prep_kernel<<<1, 256, 0, stream>>>(w1, b1, g1, be1, m1, v1, w3, b3, g2, be2,
                                     m2, v2, ALPHA, BETA, W3S, C3);
  init_a_kernel<<<(N + 255) / 256, 256, 0, stream>>>(x, A0, A1, N);
  agg_scalar_kernel<<<(E + 255) / 256, 256, 0, stream>>>(ei, x, A0, A1, E);

  for (int br = 0; br < 2; ++br) {
    const float* a = (br == 0) ? A1 : A0;  // h1 uses x[:,1], h2 uses x[:,0]
    float* H = (br == 0) ? H1 : H2;
    mlp1_kernel<<<(N + 15) / 16, 256, 0, stream>>>(a, ALPHA, BETA, w2, b2, P, N);
    hipMemcpyAsync(Q, P, (size_t)N * 128 * sizeof(float),
                   hipMemcpyDeviceToDevice, stream);
    long long tot = (long long)E * 32;
    agg128_kernel<<<(unsigned)((tot + 255) / 256), 256, 0, stream>>>(ei, P, Q, E);
    mlp2_kernel<<<(N + 15) / 16, 128, 0, stream>>>(Q, W3S, C3, H, N);
  }

  bilinear_head_kernel<<<(N + 63) / 64, 192, 0, stream>>>(
      H1, H2, wb, bb, w5, b5, w6, b6, (float*)d_out, N);
}